// AdditiveAttention_22771916603393
// MI455X (gfx1250) — compile-verified
//
#include <hip/hip_runtime.h>
#include <math.h>

typedef float v2f __attribute__((ext_vector_type(2)));
typedef float v8f __attribute__((ext_vector_type(8)));

#define BB 1024
#define SS 2048
#define HH 50
#define KP 52        // padded K: 13 steps of 4
#define NP 64        // padded N: 4 tiles of 16
#define PPSTR 68     // pp LDS row stride (avoids lo/hi half bank conflict)
#define SC 8         // s-chunks for the context pass

__device__ __forceinline__ float fast_tanh(float x) {
  float e = __expf(2.0f * x);
  return 1.0f - 2.0f * __builtin_amdgcn_rcpf(e + 1.0f);  // saturates to +/-1
}

// ---------------- Kernel A: proj_prev[b][h] = sum_k dec[b][k] * W_prev[h][k]
__global__ void pp_kernel(const float* __restrict__ dec,
                          const float* __restrict__ Wp,
                          float* __restrict__ pp) {
  int i = blockIdx.x * blockDim.x + threadIdx.x;
  if (i >= BB * HH) return;
  int b = i / HH, h = i % HH;
  const float* dr = dec + b * HH;
  const float* wr = Wp + h * HH;
  float acc = 0.f;
#pragma unroll
  for (int k = 0; k < HH; ++k) acc += dr[k] * wr[k];
  pp[i] = acc;
}

// ---------------- Kernel B: fused scores via V_WMMA_F32_16X16X4_F32
// grid = 64 b-tiles * 32 s-chunks; block = 256 (8 waves).
// Wave tile: 16 rows (16 consecutive b at fixed s) x 64 cols (h padded).
__global__ void __launch_bounds__(256)
scores_kernel(const float* __restrict__ enc,
              const float* __restrict__ Wenc,
              const float* __restrict__ Wscore,
              const float* __restrict__ pp,
              float* __restrict__ scores) {
  __shared__ float w_lds[NP * KP];        // w_lds[n*KP + k] = Wenc[n][k], zero-padded
  __shared__ float pp_lds[16 * PPSTR];    // pp for this block's 16 b-rows, zero-padded

  const int t  = threadIdx.x;
  const int bt = blockIdx.x & 63;         // which 16-b chunk
  const int sc = blockIdx.x >> 6;         // which 64-s chunk
  const int b0 = bt * 16;

  for (int i = t; i < NP * KP; i += 256) {
    int n = i / KP, k = i % KP;
    w_lds[i] = (n < HH && k < HH) ? Wenc[n * HH + k] : 0.f;
  }
  for (int i = t; i < 16 * PPSTR; i += 256) {
    int m = i / PPSTR, k = i % PPSTR;
    pp_lds[i] = (k < HH) ? pp[(b0 + m) * HH + k] : 0.f;
  }
  __syncthreads();

  const int wave = t >> 5;
  const int lane = t & 31;
  const int lm   = lane & 15;    // row (A) / col (B,C) index within 16
  const int hi   = lane >> 4;    // 0: K pair 0..1, 1: K pair 2..3

  // per-lane score weight for each n-tile (h >= 50 contributes 0)
  float wsv[4];
#pragma unroll
  for (int nt = 0; nt < 4; ++nt) {
    int h = nt * 16 + lm;
    wsv[nt] = (h < HH) ? Wscore[h] : 0.f;
  }

  const int s_base = sc * 64 + wave * 8;
#pragma clang loop unroll(disable)
  for (int it = 0; it < 8; ++it) {
    // scheduling fence: keep LDS loads inside this iteration (no LICM -> no spills)
    asm volatile("" ::: "memory");

    const int s = s_base + it;
    const float* arow = enc + ((size_t)s * BB + b0 + lm) * HH;

    // A fragments: 16x4 f32 layout (lanes 0-15: K=0,1 ; lanes 16-31: K=2,3)
    v2f a[13];
#pragma unroll
    for (int kk = 0; kk < 13; ++kk) {
      int k = kk * 4 + hi * 2;
      if (k < HH) {
        a[kk] = *(const v2f*)(arow + k);     // 8B aligned: row*50 even, k even
      } else {
        a[kk][0] = 0.f; a[kk][1] = 0.f;      // zero-pad K=50,51
      }
    }

    float rowacc[8];
#pragma unroll
    for (int v = 0; v < 8; ++v) rowacc[v] = 0.f;

#pragma unroll
    for (int nt = 0; nt < 4; ++nt) {
      // fence per n-tile: bound live B-fragments to 13 (26 VGPRs)
      asm volatile("" ::: "memory");

      v8f acc = {};
      const float* wl = &w_lds[(nt * 16 + lm) * KP + hi * 2];
#pragma unroll
      for (int kk = 0; kk < 13; ++kk) {
        v2f bfrag = *(const v2f*)(wl + kk * 4);   // B[k][n] = Wenc[n][k]
        acc = __builtin_amdgcn_wmma_f32_16x16x4_f32(
            false, a[kk], false, bfrag, (short)0, acc, false, false);
      }
      // C layout: VGPR v -> row v (lanes 0-15) / row v+8 (lanes 16-31), col = nt*16+lm
      const int h = nt * 16 + lm;
#pragma unroll
      for (int v = 0; v < 8; ++v) {
        float x = acc[v] + pp_lds[(v + 8 * hi) * PPSTR + h];
        rowacc[v] += fast_tanh(x) * wsv[nt];
      }
    }

    // per-row sum over h: reduce across the 16-lane half-groups
#pragma unroll
    for (int v = 0; v < 8; ++v) {
      float r = rowacc[v];
      r += __shfl_xor(r, 8, 16);
      r += __shfl_xor(r, 4, 16);
      r += __shfl_xor(r, 2, 16);
      r += __shfl_xor(r, 1, 16);
      rowacc[v] = r;
    }
    if (lm == 0) {
      const int rb = b0 + 8 * hi;
#pragma unroll
      for (int v = 0; v < 8; ++v)
        scores[(size_t)(rb + v) * SS + s] = rowacc[v];
    }
  }
}

// ---------------- Kernel C: masked softmax over S per batch row (in-place)
__global__ void softmax_kernel(float* __restrict__ scores,
                               const int* __restrict__ mask) {
  __shared__ float red[256];
  const int b = blockIdx.x, t = threadIdx.x;
  float x[8];
  float mx = -INFINITY;
#pragma unroll
  for (int i = 0; i < 8; ++i) {
    int s = t + i * 256;
    float v = scores[(size_t)b * SS + s];
    x[i] = (mask[(size_t)b * SS + s] != 0) ? v : -INFINITY;
    mx = fmaxf(mx, x[i]);
  }
  red[t] = mx; __syncthreads();
  for (int o = 128; o > 0; o >>= 1) {
    if (t < o) red[t] = fmaxf(red[t], red[t + o]);
    __syncthreads();
  }
  mx = red[0]; __syncthreads();
  float sum = 0.f;
#pragma unroll
  for (int i = 0; i < 8; ++i) { x[i] = __expf(x[i] - mx); sum += x[i]; }
  red[t] = sum; __syncthreads();
  for (int o = 128; o > 0; o >>= 1) {
    if (t < o) red[t] += red[t + o];
    __syncthreads();
  }
  const float inv = 1.0f / red[0];
#pragma unroll
  for (int i = 0; i < 8; ++i)
    scores[(size_t)b * SS + t + i * 256] = x[i] * inv;
}

// ---------------- Kernel D1: partial context over an S-chunk
// grid = (BB/4) * SC; block = 256 -> 64 threads per b, 256 s per chunk.
__global__ void context_partial(const float* __restrict__ enc,
                                const float* __restrict__ attn,
                                float* __restrict__ part) {
  const int t     = threadIdx.x;
  const int bblk  = blockIdx.x % (BB / 4);
  const int chunk = blockIdx.x / (BB / 4);
  const int b     = bblk * 4 + (t >> 6);
  const int h     = t & 63;
  const int s0    = chunk * (SS / SC);

  float acc = 0.f;
  if (h < HH) {
    const float* ap = attn + (size_t)b * SS + s0;
    const float* ep = enc + ((size_t)s0 * BB + b) * HH + h;
#pragma unroll 16
    for (int s = 0; s < SS / SC; ++s)
      acc = fmaf(ap[s], ep[(size_t)s * BB * HH], acc);
  }
  part[((size_t)chunk * BB + b) * 64 + h] = acc;   // h>=50 slots get 0
}

// ---------------- Kernel D2: reduce partial contexts (fixed order, deterministic)
__global__ void context_reduce(const float* __restrict__ part,
                               float* __restrict__ out) {
  int i = blockIdx.x * blockDim.x + threadIdx.x;
  if (i >= BB * HH) return;
  int b = i / HH, h = i % HH;
  float acc = 0.f;
#pragma unroll
  for (int c = 0; c < SC; ++c)
    acc += part[((size_t)c * BB + b) * 64 + h];
  out[i] = acc;
}

extern "C" void kernel_launch(void* const* d_in, const int* in_sizes, int n_in,
                              void* d_out, int out_size, void* d_ws, size_t ws_size,
                              hipStream_t stream) {
  const float* dec    = (const float*)d_in[0];  // (B,H)
  const float* enc    = (const float*)d_in[1];  // (S,B,H)
  const int*   mask   = (const int*)  d_in[2];  // (B,S)
  const float* Wp     = (const float*)d_in[3];  // (H,H)
  const float* Wenc   = (const float*)d_in[4];  // (H,H)
  const float* Wscore = (const float*)d_in[5];  // (1,H)
  float* out = (float*)d_out;                   // (B,H)

  float* pp     = (float*)d_ws;                 // B*H floats
  float* scores = pp + BB * HH;                 // B*S floats (8 MB)
  float* part   = scores + (size_t)BB * SS;     // SC*B*64 floats (2 MB)

  pp_kernel<<<(BB * HH + 255) / 256, 256, 0, stream>>>(dec, Wp, pp);
  scores_kernel<<<64 * 32, 256, 0, stream>>>(enc, Wenc, Wscore, pp, scores);
  softmax_kernel<<<BB, 256, 0, stream>>>(scores, mask);
  context_partial<<<(BB / 4) * SC, 256, 0, stream>>>(enc, scores, part);
  context_reduce<<<(BB * HH + 255) / 256, 256, 0, stream>>>(part, out);
}